// CSC_10058813407511
// MI455X (gfx1250) — compile-verified
//
#include <hip/hip_runtime.h>

// CSC forward for MI455X (gfx1250), wave32.
// Integral image built with V_WMMA_F32_16X16X4_F32 via triangular-ones matmuls.

#define N_IMG   4
#define N_CLS   20
#define H_IMG   512
#define W_IMG   512
#define N_ROI   4000
#define MAPS    (N_IMG * N_CLS)          // 80
#define MAP_EL  (H_IMG * W_IMG)          // 262144
#define RC      (N_ROI * N_CLS)          // 80000
#define EPSF    1e-12f
#define NEG_BIG (-3.402823466e38f)

typedef __attribute__((ext_vector_type(2))) float v2f;
typedef __attribute__((ext_vector_type(8))) float v8f;

// ---------------------------------------------------------------- K1: per-map max
__global__ void k_mapmax(const float* __restrict__ cpgs, float* __restrict__ mx) {
    __shared__ float red[256];
    const int nm  = blockIdx.x;
    const int tid = threadIdx.x;
    const float* p = cpgs + (size_t)nm * MAP_EL;
    float m = NEG_BIG;
    for (int i = tid; i < MAP_EL; i += 256) m = fmaxf(m, p[i]);
    red[tid] = m;
    __syncthreads();
    for (int s = 128; s > 0; s >>= 1) {
        if (tid < s) red[tid] = fmaxf(red[tid], red[tid + s]);
        __syncthreads();
    }
    if (tid == 0) mx[nm] = red[0];
}

// ---------------------------------------------------------------- K2: threshold + row prefix
// One wave per 16-row stripe. Per 16x16 tile: D = X * U + carry, via 4 chained
// wmma_f32_16x16x4. U = upper-triangular ones (constant, built in B layout).
__global__ void k_rowprefix(const float* __restrict__ cpgs,
                            const float* __restrict__ mx,
                            float* __restrict__ ii) {
    const int wave   = (blockIdx.x * blockDim.x + threadIdx.x) >> 5; // 0..2559
    const int lane   = threadIdx.x & 31;
    const int nm     = wave >> 5;          // 32 stripes per map
    const int stripe = wave & 31;
    const int l15    = lane & 15;
    const int hi     = lane >> 4;
    const int kbase  = hi * 2;
    const float thr  = 0.1f * mx[nm];

    const float* __restrict__ src = cpgs + (size_t)nm * MAP_EL;
    float* __restrict__ dst       = ii   + (size_t)nm * MAP_EL;
    const int rb = stripe * 16;

    // B = U (4x16 per chunk): U[k][n] = (k <= n)
    v2f Bk[4];
#pragma unroll
    for (int c = 0; c < 4; ++c) {
        Bk[c].x = ((4 * c + kbase)     <= l15) ? 1.0f : 0.0f;
        Bk[c].y = ((4 * c + kbase + 1) <= l15) ? 1.0f : 0.0f;
    }

    v8f carry = {0.f, 0.f, 0.f, 0.f, 0.f, 0.f, 0.f, 0.f};

    for (int t = 0; t < 32; ++t) {
        const int cb = t * 16;
        v8f acc = carry;                      // C[m][n] = rowcarry[m]
#pragma unroll
        for (int c = 0; c < 4; ++c) {
            const float* p = src + (size_t)(rb + l15) * W_IMG + cb + 4 * c + kbase;
            const float x0 = p[0];
            const float x1 = p[1];
            v2f A;
            A.x = (x0 >= thr) ? x0 : 0.0f;    // fused FG threshold
            A.y = (x1 >= thr) ? x1 : 0.0f;
            acc = __builtin_amdgcn_wmma_f32_16x16x4_f32(
                false, A, false, Bk[c], (short)0, acc, false, false);
        }
        // store D: lane holds rows v + 8*hi, col = l15
#pragma unroll
        for (int v = 0; v < 8; ++v)
            dst[(size_t)(rb + v + 8 * hi) * W_IMG + cb + l15] = acc[v];
        // new carry[row] = D[row][15]  (col 15 lives in lanes 15 / 31)
#pragma unroll
        for (int v = 0; v < 8; ++v)
            carry[v] = __shfl(acc[v], hi ? 31 : 15, 32);
    }
}

// ---------------------------------------------------------------- K3: column prefix (in place)
// One wave per 16-col stripe. Per 16x16 tile: D = L * X + carry.
// L = lower-triangular ones (constant, built in A layout).
__global__ void k_colprefix(float* __restrict__ ii) {
    const int wave   = (blockIdx.x * blockDim.x + threadIdx.x) >> 5;
    const int lane   = threadIdx.x & 31;
    const int nm     = wave >> 5;
    const int stripe = wave & 31;
    const int l15    = lane & 15;
    const int hi     = lane >> 4;
    const int kbase  = hi * 2;

    float* __restrict__ map = ii + (size_t)nm * MAP_EL;
    const int cb = stripe * 16;

    // A = L (16x4 per chunk): L[m][k] = (k <= m)
    v2f Ak[4];
#pragma unroll
    for (int c = 0; c < 4; ++c) {
        Ak[c].x = ((4 * c + kbase)     <= l15) ? 1.0f : 0.0f;
        Ak[c].y = ((4 * c + kbase + 1) <= l15) ? 1.0f : 0.0f;
    }

    float cc = 0.0f;                          // per-column running sum

    for (int t = 0; t < 32; ++t) {
        const int rb = t * 16;
        v8f acc = {cc, cc, cc, cc, cc, cc, cc, cc};
#pragma unroll
        for (int c = 0; c < 4; ++c) {
            v2f B;                            // B[k][n] = X[rb + k][cb + n]
            B.x = map[(size_t)(rb + 4 * c + kbase)     * W_IMG + cb + l15];
            B.y = map[(size_t)(rb + 4 * c + kbase + 1) * W_IMG + cb + l15];
            acc = __builtin_amdgcn_wmma_f32_16x16x4_f32(
                false, Ak[c], false, B, (short)0, acc, false, false);
        }
#pragma unroll
        for (int v = 0; v < 8; ++v)
            map[(size_t)(rb + v + 8 * hi) * W_IMG + cb + l15] = acc[v];
        // new carry[col] = D[15][col] = acc[7] of lanes 16..31
        cc = __shfl(acc[7], 16 + l15, 32);
    }
}

// ---------------------------------------------------------------- K4: per-(ROI,class) score
__device__ __forceinline__ float g_lookup(const float* __restrict__ S, int y, int x) {
    // padded integral image lookup: ii_pad[y][x] == S[y-1][x-1], zero on borders
    return (y > 0 && x > 0) ? S[(size_t)(y - 1) * W_IMG + (x - 1)] : 0.0f;
}
__device__ __forceinline__ float boxsum(const float* __restrict__ S,
                                        int x1, int y1, int x2, int y2) {
    return g_lookup(S, y2 + 1, x2 + 1) - g_lookup(S, y1, x2 + 1)
         - g_lookup(S, y2 + 1, x1)     + g_lookup(S, y1, x1);
}
__device__ __forceinline__ int clampRound(float v, float lo, float hi) {
    return (int)fminf(fmaxf(rintf(v), lo), hi);   // rintf = round half-to-even (jnp.round)
}

__global__ void k_score(const float* __restrict__ rois,
                        const float* __restrict__ ii,
                        float* __restrict__ score,
                        float* __restrict__ validf) {
    const int idx = blockIdx.x * blockDim.x + threadIdx.x;
    if (idx >= RC) return;
    const int r = idx / N_CLS;
    const int c = idx % N_CLS;
    const float* rp = rois + (size_t)r * 5;
    const int b  = (int)rp[0];
    const int x1 = clampRound(rp[1], 0.f, (float)(W_IMG - 1));
    const int y1 = clampRound(rp[2], 0.f, (float)(H_IMG - 1));
    const int x2 = clampRound(rp[3], 0.f, (float)(W_IMG - 1));
    const int y2 = clampRound(rp[4], 0.f, (float)(H_IMG - 1));

    const float* __restrict__ S = ii + (size_t)(b * N_CLS + c) * MAP_EL;

    const float s_in = boxsum(S, x1, y1, x2, y2);
    const float a_in = (float)((x2 - x1 + 1) * (y2 - y1 + 1));

    const float cx = (float)(x1 + x2) * 0.5f;
    const float cy = (float)(y1 + y2) * 0.5f;
    const float hw = (float)(x2 - x1 + 1) * 0.9f;   // CONTEXT_SCALE * 0.5
    const float hh = (float)(y2 - y1 + 1) * 0.9f;
    const int cx1 = clampRound(cx - hw, 0.f, (float)(W_IMG - 1));
    const int cy1 = clampRound(cy - hh, 0.f, (float)(H_IMG - 1));
    const int cx2 = clampRound(cx + hw, 0.f, (float)(W_IMG - 1));
    const int cy2 = clampRound(cy + hh, 0.f, (float)(H_IMG - 1));

    const float s_ctx = boxsum(S, cx1, cy1, cx2, cy2);
    const float a_ctx = (float)((cx2 - cx1 + 1) * (cy2 - cy1 + 1));
    const float s_fr  = s_ctx - s_in;
    const float a_fr  = fmaxf(a_ctx - a_in, 1.0f);

    const float sc = s_in / sqrtf(a_in) - s_fr / sqrtf(a_fr);
    score[idx] = sc;

    const float tot   = S[(size_t)(H_IMG - 1) * W_IMG + (W_IMG - 1)]; // m.sum()
    const float massf = s_in / fmaxf(tot, EPSF);
    const float dens  = (s_in / a_in) / fmaxf(tot / (float)(H_IMG * W_IMG), EPSF);
    validf[idx] = (massf >= 0.2f && dens >= 0.0f) ? 1.0f : 0.0f;
}

// ---------------------------------------------------------------- K5: segment max pos/neg
__global__ void k_posneg(const float* __restrict__ rois,
                         const float* __restrict__ score,
                         float* __restrict__ pos,
                         float* __restrict__ neg) {
    __shared__ float rp[256];
    __shared__ float rn[256];
    const int nm  = blockIdx.x;           // (n, c)
    const int n   = nm / N_CLS;
    const int c   = nm % N_CLS;
    const int tid = threadIdx.x;
    float mp = NEG_BIG, mn = NEG_BIG;
    for (int r = tid; r < N_ROI; r += 256) {
        if ((int)rois[(size_t)r * 5] == n) {
            const float s = score[(size_t)r * N_CLS + c];
            mp = fmaxf(mp, s);
            mn = fmaxf(mn, -s);
        }
    }
    rp[tid] = mp; rn[tid] = mn;
    __syncthreads();
    for (int s = 128; s > 0; s >>= 1) {
        if (tid < s) {
            rp[tid] = fmaxf(rp[tid], rp[tid + s]);
            rn[tid] = fmaxf(rn[tid], rn[tid + s]);
        }
        __syncthreads();
    }
    if (tid == 0) { pos[nm] = rp[0]; neg[nm] = rn[0]; }
}

// ---------------------------------------------------------------- K6: normalize, gate, write out
__global__ void k_final(const float* __restrict__ rois,
                        const float* __restrict__ score,
                        const float* __restrict__ validf,
                        const float* __restrict__ labels,
                        const float* __restrict__ preds,
                        const float* __restrict__ pos,
                        const float* __restrict__ neg,
                        float* __restrict__ out) {
    const int idx = blockIdx.x * blockDim.x + threadIdx.x;
    if (idx < MAPS) {                      // PL = labels, NL = zeros
        out[RC + idx]        = labels[idx];
        out[RC + MAPS + idx] = 0.0f;
    }
    if (idx >= RC) return;
    const int r = idx / N_CLS;
    const int c = idx % N_CLS;
    const int b = (int)rois[(size_t)r * 5];
    const int nc = b * N_CLS + c;

    const float s = score[idx];
    float w = (s >= 0.0f) ? s / fmaxf(pos[nc], EPSF)
                          : s / fmaxf(neg[nc], EPSF);
    if (validf[idx] == 0.0f) w = 0.0f;
    const bool active = (labels[nc] >= 0.5f) && (preds[nc] >= 0.7f);
    out[idx] = active ? fminf(fmaxf(w, -1.0f), 1.0f) : 1.0f;
}

// ---------------------------------------------------------------- launch
extern "C" void kernel_launch(void* const* d_in, const int* in_sizes, int n_in,
                              void* d_out, int out_size, void* d_ws, size_t ws_size,
                              hipStream_t stream) {
    const float* cpgs   = (const float*)d_in[0];
    const float* labels = (const float*)d_in[1];
    const float* preds  = (const float*)d_in[2];
    const float* rois   = (const float*)d_in[3];
    float* out = (float*)d_out;

    // workspace layout (floats)
    float* ws     = (float*)d_ws;
    float* ii     = ws;                               // 80 * 512 * 512
    float* mx     = ii + (size_t)MAPS * MAP_EL;       // 80
    float* score  = mx + MAPS;                        // 80000
    float* validf = score + RC;                       // 80000
    float* pos    = validf + RC;                      // 80
    float* neg    = pos + MAPS;                       // 80

    // K1: per-map max
    k_mapmax<<<MAPS, 256, 0, stream>>>(cpgs, mx);

    // K2/K3: integral image via WMMA prefix-sum passes.
    // 80 maps * 32 stripes = 2560 waves; 4 waves (128 threads) per block.
    k_rowprefix<<<640, 128, 0, stream>>>(cpgs, mx, ii);
    k_colprefix<<<640, 128, 0, stream>>>(ii);

    // K4: per-(ROI,class) box sums + scores (+ validity gate)
    k_score<<<(RC + 255) / 256, 256, 0, stream>>>(rois, ii, score, validf);

    // K5: per-(image,class) segment max of +/- score
    k_posneg<<<MAPS, 256, 0, stream>>>(rois, score, pos, neg);

    // K6: normalize, gate, emit W / PL / NL
    k_final<<<(RC + 255) / 256, 256, 0, stream>>>(rois, score, validf, labels,
                                                  preds, pos, neg, out);
}